// FromRGB_65592740544928
// MI455X (gfx1250) — compile-verified
//
#include <hip/hip_runtime.h>

typedef float v2f __attribute__((ext_vector_type(2)));
typedef float v8f __attribute__((ext_vector_type(8)));

#define LRELU_SLOPE 0.2f
#define FUSED_SCALE 1.41421356237309515f   // sqrt(2)
#define WSCALE 0.28867513459481287f        // 1/sqrt(12)

// ---------------------------------------------------------------------------
// Kernel 1: fused inverse-Haar(up2) -> [1,3,3,1]^2/64 blur(down2) -> Haar(down2)
// x: (B,12,256,256)  ->  t: (B,12,128,128), channel layout [ll|lh|hl|hh] x 3
// Each block: 16x16 output tile for one (b, cg). LDS holds the 34x34 x-window
// of all 4 subband groups (zero-padded; this reproduces all upfirdn padding).
// ---------------------------------------------------------------------------
__global__ __launch_bounds__(256) void wavelet_fused(const float* __restrict__ x,
                                                     float* __restrict__ t) {
  __shared__ float lds[4 * 34 * 36];
  const int tile = blockIdx.x;   // 0..63 (8x8 tiles of 16x16)
  const int cg   = blockIdx.y;   // 0..2  channel within group
  const int b    = blockIdx.z;
  const int u0 = (tile >> 3) << 4;
  const int v0 = (tile & 7) << 4;
  const int tid = threadIdx.x;

  // x rows/cols needed: [2*u0-1, 2*u0+32], same for cols (34 x 34), 4 groups
  const int row0 = 2 * u0 - 1;
  const int col0 = 2 * v0 - 1;
  for (int idx = tid; idx < 4 * 34 * 34; idx += 256) {
    int g   = idx / (34 * 34);
    int rem = idx - g * (34 * 34);
    int lr  = rem / 34;
    int lc  = rem - lr * 34;
    int r = row0 + lr;
    int c = col0 + lc;
    float val = 0.0f;
    if (r >= 0 && r < 256 && c >= 0 && c < 256)
      val = x[(((size_t)(b * 12 + g * 3 + cg) * 256) + r) * 256 + c];
    lds[(g * 34 + lr) * 36 + lc] = val;
  }
  __syncthreads();

  const int ty = tid >> 4;
  const int tx = tid & 15;
  const int u = u0 + ty;
  const int v = v0 + tx;
  const int lrb = 2 * ty;   // local x-row base (x row 2u-1)
  const int lcb = 2 * tx;

  // Inverse-Haar combos at the 4x4 x window:
  // y512[2m+dy, 2n+dx] = 0.5 * combo_{dy,dx}(m,n)
  float c00[4][4], c01[4][4], c10[4][4], c11[4][4];
#pragma unroll
  for (int i = 0; i < 4; ++i) {
#pragma unroll
    for (int j = 0; j < 4; ++j) {
      float a  = lds[(0 * 34 + lrb + i) * 36 + lcb + j];
      float bb = lds[(1 * 34 + lrb + i) * 36 + lcb + j];
      float cc = lds[(2 * 34 + lrb + i) * 36 + lcb + j];
      float dd = lds[(3 * 34 + lrb + i) * 36 + lcb + j];
      float s0 = a + bb, s1 = cc + dd, s2 = a - bb, s3 = cc - dd;
      c00[i][j] = 0.5f * (s0 + s1);
      c01[i][j] = 0.5f * (s0 - s1);
      c10[i][j] = 0.5f * (s2 + s3);
      c11[i][j] = 0.5f * (s2 - s3);
    }
  }

  const float w4[4] = {1.0f, 3.0f, 3.0f, 1.0f};

  // y512 rows p = 4u-1+i, i=0..5: x-row (i+1)>>1, parity (i+1)&1.
  // Horizontal blur over y512 cols q = 4v-1+j, j=0..5.
  float hs[6][2];
#pragma unroll
  for (int i = 0; i < 6; ++i) {
    const int lrow = (i + 1) >> 1;
    const int dy   = (i + 1) & 1;
    float yv[6];
#pragma unroll
    for (int j = 0; j < 6; ++j) {
      const int lcol = (j + 1) >> 1;
      const int dx   = (j + 1) & 1;
      float vv;
      if (dy == 0) vv = (dx == 0) ? c00[lrow][lcol] : c01[lrow][lcol];
      else         vv = (dx == 0) ? c10[lrow][lcol] : c11[lrow][lcol];
      yv[j] = vv;
    }
#pragma unroll
    for (int sc = 0; sc < 2; ++sc)
      hs[i][sc] = w4[0] * yv[2 * sc] + w4[1] * yv[2 * sc + 1] +
                  w4[2] * yv[2 * sc + 2] + w4[3] * yv[2 * sc + 3];
  }

  // Vertical blur -> 2x2 patch of the 256x256 intermediate
  float y2[2][2];
#pragma unroll
  for (int rc = 0; rc < 2; ++rc)
#pragma unroll
    for (int sc = 0; sc < 2; ++sc)
      y2[rc][sc] = (1.0f / 64.0f) *
          (w4[0] * hs[2 * rc][sc] + w4[1] * hs[2 * rc + 1][sc] +
           w4[2] * hs[2 * rc + 2][sc] + w4[3] * hs[2 * rc + 3][sc]);

  const float y00 = y2[0][0], y01 = y2[0][1], y10 = y2[1][0], y11 = y2[1][1];
  const float tll = 0.5f * (y00 + y01 + y10 + y11);
  const float tlh = 0.5f * (y00 + y01 - y10 - y11);
  const float thl = 0.5f * (y00 - y01 + y10 - y11);
  const float thh = 0.5f * (y00 - y01 - y10 + y11);

  const size_t plane = 128 * 128;
  const size_t pix = (size_t)u * 128 + v;
  float* tb = t + (size_t)b * 12 * plane + pix;
  tb[(size_t)(0 * 3 + cg) * plane] = tll;
  tb[(size_t)(1 * 3 + cg) * plane] = tlh;
  tb[(size_t)(2 * 3 + cg) * plane] = thl;
  tb[(size_t)(3 * 3 + cg) * plane] = thh;
}

// ---------------------------------------------------------------------------
// Kernel 2: out[b,o,p] = lrelu( sum_c t[b,c,p] * W[o,c]/sqrt(12) + bias[o] ) * sqrt(2)
// One wave owns a 16-pixel strip: load the A operand (t tile, K=12) ONCE, then
// loop over all 16 output-channel tiles (3x V_WMMA_F32_16X16X4_F32 each).
// This reads t exactly once from HBM; weight (12 KB) stays cache-resident.
// A layout (16x4 f32): lanes 0-15 M=lane, VGPR0/1 = K0/K1; lanes 16-31 = K2/K3.
// D layout: lane -> N, VGPR r -> M=r (lanes 0-15) / M=8+r (lanes 16-31).
// ---------------------------------------------------------------------------
__global__ __launch_bounds__(256) void gemm_wmma(const float* __restrict__ t,
                                                 const float* __restrict__ w,
                                                 const float* __restrict__ bias,
                                                 float* __restrict__ out) {
  const int lane = threadIdx.x & 31;
  const int wv   = threadIdx.x >> 5;
  const int b    = blockIdx.x >> 7;                       // 128 blocks per batch
  const int pixbase = ((blockIdx.x & 127) * 8 + wv) * 16; // 16 pixels per wave
  const int half = lane >> 4;
  const int mn   = lane & 15;

  // A operand: 16 pixels x K=12 of t, loaded once, held in 6 VGPRs.
  const float* tb = t + (size_t)b * 12 * 16384;
  v2f am[3];
#pragma unroll
  for (int kg = 0; kg < 3; ++kg) {
    const int k0 = kg * 4 + half * 2;
    am[kg][0] = tb[(size_t)k0 * 16384 + pixbase + mn];
    am[kg][1] = tb[(size_t)(k0 + 1) * 16384 + pixbase + mn];
  }

  // lane holds 8 consecutive pixels (m = half*8 + r) at fixed o
  float* ob = out + (size_t)b * 256 * 16384 + pixbase + half * 8;

  for (int nt = 0; nt < 16; ++nt) {
    const int o = nt * 16 + mn;

    v8f acc = {0.f, 0.f, 0.f, 0.f, 0.f, 0.f, 0.f, 0.f};
#pragma unroll
    for (int kg = 0; kg < 3; ++kg) {
      const int k0 = kg * 4 + half * 2;
      v2f bm;
      bm[0] = w[o * 12 + k0] * WSCALE;
      bm[1] = w[o * 12 + k0 + 1] * WSCALE;
      acc = __builtin_amdgcn_wmma_f32_16x16x4_f32(
          /*neg_a=*/false, am[kg], /*neg_b=*/false, bm,
          /*c_mod=*/(short)0, acc, /*reuse_a=*/false, /*reuse_b=*/false);
    }

    const float bv = bias[o];
    float res[8];
#pragma unroll
    for (int r = 0; r < 8; ++r) {
      float vv = acc[r] + bv;
      vv = (vv > 0.0f) ? vv : (LRELU_SLOPE * vv);
      res[r] = vv * FUSED_SCALE;
    }
    float* op = ob + (size_t)o * 16384;
    *(float4*)(op)     = make_float4(res[0], res[1], res[2], res[3]);
    *(float4*)(op + 4) = make_float4(res[4], res[5], res[6], res[7]);
  }
}

extern "C" void kernel_launch(void* const* d_in, const int* in_sizes, int n_in,
                              void* d_out, int out_size, void* d_ws, size_t ws_size,
                              hipStream_t stream) {
  const float* x      = (const float*)d_in[0];  // (B,12,256,256)
  const float* weight = (const float*)d_in[1];  // (256,12,1,1)
  const float* bias   = (const float*)d_in[2];  // (256,)
  const int B = in_sizes[0] / (12 * 256 * 256);

  float* t   = (float*)d_out;                               // (B,12,128,128)
  float* out = (float*)d_out + (size_t)B * 12 * 128 * 128;  // (B,256,128,128)

  dim3 g1(64, 3, B);
  wavelet_fused<<<g1, 256, 0, stream>>>(x, t);

  dim3 g2(B * 128, 1, 1);
  gemm_wmma<<<g2, 256, 0, stream>>>(t, weight, bias, out);
}